// Net_23776938951392
// MI455X (gfx1250) — compile-verified
//
#include <hip/hip_runtime.h>

typedef float v2f __attribute__((ext_vector_type(2)));
typedef float v8f __attribute__((ext_vector_type(8)));

#define CIN   16
#define COUTC 16
#define IMG_H 1024
#define IMG_W 1024
#define TH    16      // output tile rows per workgroup
#define TW    32      // output tile cols per workgroup
#define HALO_Y (TH + 2)
#define HALO_X (TW + 2)
#define CPITCH 18     // 16 cin + 2 pad dwords: even (b64 align) + bank spread

#if defined(__gfx1250__) && __has_builtin(__builtin_amdgcn_global_load_async_to_lds_b32)
#define USE_ASYNC_STAGE 1
#else
#define USE_ASYNC_STAGE 0
#endif

typedef __attribute__((address_space(1))) int* gas_i32;   // global int*
typedef __attribute__((address_space(3))) int* las_i32;   // LDS int*

__device__ __forceinline__ void stage_b32(const float* g, float* l) {
#if USE_ASYNC_STAGE
  __builtin_amdgcn_global_load_async_to_lds_b32(
      (gas_i32)const_cast<float*>(g), (las_i32)l,
      /*offset=*/0, /*cpol=*/0);
#else
  *l = *g;
#endif
}

__device__ __forceinline__ void drain_async() {
#if USE_ASYNC_STAGE
#if __has_builtin(__builtin_amdgcn_s_wait_asynccnt)
  __builtin_amdgcn_s_wait_asynccnt(0);
#else
  asm volatile("s_wait_asynccnt 0x0" ::: "memory");
#endif
#endif
}

__launch_bounds__(256)
__global__ void conv3x3_wmma_f32(const float* __restrict__ X,
                                 const float* __restrict__ Wt,
                                 float* __restrict__ Out) {
  // input tile with halo: [y][x][cin(+pad)]
  __shared__ __attribute__((aligned(16))) float sIn[HALO_Y * HALO_X * CPITCH];
  // weights: [k2][m][j] with k = 2*k2 + j, k = (kh*3+kw)*16 + cin
  __shared__ __attribute__((aligned(16))) float sW[72 * 32];

  const int tid = threadIdx.x;
  const int w0  = blockIdx.x * TW;
  const int h0  = blockIdx.y * TH;
  const int b   = blockIdx.z;

  // ---- stage weights (OIHW -> [k2][m][j]) directly into LDS ----
  for (int t = tid; t < 72 * 32; t += 256) {
    int k2  = t >> 5;
    int rem = t & 31;
    int m   = rem >> 1;
    int j   = rem & 1;
    int k   = k2 * 2 + j;
    int s   = k >> 4;        // kh*3 + kw
    int cin = k & 15;
    int kh  = s / 3;
    int kw  = s % 3;
    stage_b32(&Wt[((m * CIN + cin) * 3 + kh) * 3 + kw], &sW[t]);
  }

  // ---- stage haloed input tile, x fastest for coalesced global reads ----
  const float* Xb = X + (size_t)b * CIN * IMG_H * IMG_W;
  for (int t = tid; t < HALO_Y * CIN * HALO_X; t += 256) {
    int x   = t % HALO_X;
    int r   = t / HALO_X;
    int cin = r & 15;
    int y   = r >> 4;
    int gh  = h0 + y - 1;
    int gw  = w0 + x - 1;
    float* l = &sIn[(y * HALO_X + x) * CPITCH + cin];
    if ((unsigned)gh < IMG_H && (unsigned)gw < IMG_W) {
      stage_b32(&Xb[(size_t)cin * IMG_H * IMG_W + (size_t)gh * IMG_W + gw], l);
    } else {
      *l = 0.0f;   // halo zero-fill via DS store
    }
  }

  drain_async();     // ASYNCcnt -> 0 before the barrier makes LDS visible
  __syncthreads();

  const int wv   = tid >> 5;   // wave id 0..7
  const int lane = tid & 31;
  const int n    = lane & 15;  // = N (pixel) for B/C/D, = M (cout) for A
  const int half = lane >> 4;  // lane group: selects K offset {0,1} vs {2,3}

  // each wave: 2 output rows x 2 column groups of 16 pixels
  for (int g = 0; g < 4; ++g) {
    const int y  = wv * 2 + (g >> 1);
    const int xg = (g & 1) * 16;

    v8f acc = {};
#pragma unroll
    for (int ks = 0; ks < 36; ++ks) {       // K = 144 in steps of 4
      const int s    = ks >> 2;             // kh*3+kw (const per 4 steps)
      const int kh   = s / 3;
      const int kw   = s % 3;
      const int cin0 = ((ks & 3) << 2) + half * 2;
      const int k2   = ks * 2 + half;

      // A fragment: W rows M=0..15, K = 4*ks + {0,1}(lanes 0-15) / {2,3}(16-31)
      v2f a = *(const v2f*)&sW[k2 * 32 + n * 2];
      // B fragment: im2col, N = pixel, same K split per lane half
      v2f bb = *(const v2f*)&sIn[((y + kh) * HALO_X + (xg + n + kw)) * CPITCH + cin0];

      acc = __builtin_amdgcn_wmma_f32_16x16x4_f32(false, a, false, bb,
                                                  (short)0, acc, false, false);
    }

    // C/D layout: VGPR v holds M = v + 8*half, lane n = N
    const int h = h0 + y;
    const int w = w0 + xg + n;
#pragma unroll
    for (int v = 0; v < 8; ++v) {
      int cout = v + half * 8;
      Out[(((size_t)b * COUTC + cout) * IMG_H + h) * IMG_W + w] = acc[v];
    }
  }
}

extern "C" void kernel_launch(void* const* d_in, const int* in_sizes, int n_in,
                              void* d_out, int out_size, void* d_ws, size_t ws_size,
                              hipStream_t stream) {
  const float* x  = (const float*)d_in[0];   // (8,16,1024,1024) fp32
  const float* Wt = (const float*)d_in[1];   // (16,16,3,3) fp32, OIHW
  float* out      = (float*)d_out;           // (8,16,1024,1024) fp32

  dim3 grid(IMG_W / TW, IMG_H / TH, 8);
  dim3 block(256);
  conv3x3_wmma_f32<<<grid, block, 0, stream>>>(x, Wt, out);
}